// yolov8_GradCamLoss_box_49400713839279
// MI455X (gfx1250) — compile-verified
//
#include <hip/hip_runtime.h>

typedef float v2f __attribute__((ext_vector_type(2)));
typedef float v8f __attribute__((ext_vector_type(8)));

#define T_DIM 16
#define H_DIM 1280
#define W_DIM 1280
#define NBOX  64
#define TILE_Y 16
#define TILE_X 64
#define LDS_STRIDE (TILE_X + 4)   // 68 floats: WMMA-layout reads hit banks (4*row+x)%64 -> conflict-free

// ---------------------------------------------------------------------------
// Main kernel: one block = one 16x64 tile of the t-summed image.
// Streams data once (bandwidth bound), contracts against per-box column
// indicators with v_wmma_f32_16x16x4_f32, applies row indicators in epilogue.
// ---------------------------------------------------------------------------
__global__ __launch_bounds__(256)
void gradcam_box_main(const float* __restrict__ data,
                      const int*   __restrict__ boxes,
                      float*       __restrict__ partials)
{
    __shared__ float Mt[TILE_Y * LDS_STRIDE];
    __shared__ int4  sbox[NBOX];
    __shared__ float red[256];

    const int tid = threadIdx.x;
    if (tid < NBOX) sbox[tid] = ((const int4*)boxes)[tid];   // {x1,y1,x2,y2}

    const int y0 = blockIdx.x * TILE_Y;
    const int x0 = blockIdx.y * TILE_X;

    // ---- stage: M[row][x] = sum_t data[t, y0+row, x0+x], coalesced b128 loads
    const int row = tid >> 4;          // 0..15
    const int c4  = tid & 15;          // 0..15 (float4 column)
    float4 acc = make_float4(0.f, 0.f, 0.f, 0.f);
    size_t base = (size_t)(y0 + row) * W_DIM + (size_t)(x0 + c4 * 4);
    const size_t plane = (size_t)H_DIM * W_DIM;
#pragma unroll
    for (int t = 0; t < T_DIM; ++t) {
        float4 v = *(const float4*)(data + base + (size_t)t * plane);
        acc.x += v.x; acc.y += v.y; acc.z += v.z; acc.w += v.w;
    }
    float* mrow = &Mt[row * LDS_STRIDE + c4 * 4];
    mrow[0] = acc.x; mrow[1] = acc.y; mrow[2] = acc.z; mrow[3] = acc.w;
    __syncthreads();

    // ---- WMMA contraction: G[y, box] += sum_x M[y,x] * colind(x, box)
    const int wave = tid >> 5;   // 0..7, each wave owns 2 K-chunks of 4 cols
    const int lane = tid & 31;
    const int l16  = lane & 15;
    const int hi   = lane >> 4;  // 0 or 1 (lane half)

    v8f cacc[4] = { v8f{}, v8f{}, v8f{}, v8f{} };

#pragma unroll
    for (int cc = 0; cc < 2; ++cc) {
        const int k0 = (wave * 2 + cc) * 4;
        // A operand: f32 16x4 layout -> lanes0-15: K={k0,k0+1}, lanes16-31: K={k0+2,k0+3}
        v2f a;
        a.x = Mt[l16 * LDS_STRIDE + k0 + 2 * hi];
        a.y = Mt[l16 * LDS_STRIDE + k0 + 2 * hi + 1];
        const int xg = x0 + k0 + 2 * hi;   // global column for this lane's K pair
#pragma unroll
        for (int j = 0; j < 4; ++j) {      // 4 box tiles of 16
            const int4 bx = sbox[j * 16 + l16];
            v2f b;
            b.x = (xg     >= bx.x && xg     < bx.z) ? 1.0f : 0.0f;
            b.y = (xg + 1 >= bx.x && xg + 1 < bx.z) ? 1.0f : 0.0f;
            cacc[j] = __builtin_amdgcn_wmma_f32_16x16x4_f32(
                false, a, false, b, (short)0, cacc[j], false, false);
        }
    }

    // ---- epilogue: weight D by row indicators.
    // C/D layout: VGPR r -> M = r (lanes0-15) / r+8 (lanes16-31), N = lane&15
    float local = 0.f;
#pragma unroll
    for (int j = 0; j < 4; ++j) {
        const int4 bx = sbox[j * 16 + l16];
#pragma unroll
        for (int r = 0; r < 8; ++r) {
            const int y = y0 + r + 8 * hi;
            const float w = (y >= bx.y && y < bx.w) ? 1.0f : 0.0f;
            local += w * cacc[j][r];
        }
    }

    red[tid] = local;
    __syncthreads();
#pragma unroll
    for (int s = 128; s > 0; s >>= 1) {
        if (tid < s) red[tid] += red[tid + s];
        __syncthreads();
    }
    if (tid == 0)
        partials[blockIdx.y * gridDim.x + blockIdx.x] = red[0];
}

// ---------------------------------------------------------------------------
// Final reduction: 1600 partials -> scalar (deterministic, atomic-free)
// ---------------------------------------------------------------------------
__global__ __launch_bounds__(256)
void gradcam_box_final(const float* __restrict__ partials,
                       float* __restrict__ out, int n)
{
    __shared__ float red[256];
    float s = 0.f;
    for (int i = threadIdx.x; i < n; i += 256) s += partials[i];
    red[threadIdx.x] = s;
    __syncthreads();
#pragma unroll
    for (int k = 128; k > 0; k >>= 1) {
        if (threadIdx.x < k) red[threadIdx.x] += red[threadIdx.x + k];
        __syncthreads();
    }
    if (threadIdx.x == 0) out[0] = red[0];
}

extern "C" void kernel_launch(void* const* d_in, const int* in_sizes, int n_in,
                              void* d_out, int out_size, void* d_ws, size_t ws_size,
                              hipStream_t stream)
{
    (void)in_sizes; (void)n_in; (void)out_size; (void)ws_size;
    const float* data  = (const float*)d_in[0];   // (16,1280,1280) f32
    const int*   boxes = (const int*)d_in[1];     // (64,4) i32
    float* partials = (float*)d_ws;               // 1600 floats of scratch
    float* out      = (float*)d_out;

    dim3 grid(H_DIM / TILE_Y, W_DIM / TILE_X);    // 80 x 20 = 1600 blocks
    gradcam_box_main<<<grid, 256, 0, stream>>>(data, boxes, partials);
    gradcam_box_final<<<1, 256, 0, stream>>>(partials, out,
                                             (H_DIM / TILE_Y) * (W_DIM / TILE_X));
}